// MultiHeadAttention_1881195676299
// MI455X (gfx1250) — compile-verified
//
#include <hip/hip_runtime.h>

typedef float v2f __attribute__((ext_vector_type(2)));
typedef float v8f __attribute__((ext_vector_type(8)));

#define D_MODEL 768
#define NHEAD   12
#define DK      64
#define BATCH   2
#define SEQ     2048
#define M_ROWS  (BATCH * SEQ)   /* 4096 */
#define NEGV    (-1000000000.0f)
#define SMSTRIDE 18             /* even -> 8B-aligned float2, conflict-free banks */
#define WPB      8              /* waves per block */

// ---------------------------------------------------------------------------
// C[M,768] = A[M,768] @ W[768,768] + bias ; optional head-split output layout.
// One wave32 computes a 16(M) x 64(N) tile: the A float2 load is shared
// across 4 N-chunks -> 2.25 memory ops per WMMA instead of 4.
// ---------------------------------------------------------------------------
__global__ __launch_bounds__(256) void gemm768_kernel(
    const float* __restrict__ A, const float* __restrict__ W,
    const float* __restrict__ bias, float* __restrict__ C, int split_heads)
{
    int wave = (int)((blockIdx.x * blockDim.x + threadIdx.x) >> 5);
    int lane = (int)(threadIdx.x & 31);
    int tn = wave % (D_MODEL / 64);          /* 12 column groups of 64 */
    int tm = wave / (D_MODEL / 64);
    int m0 = tm * 16, n0 = tn * 64;
    int l15 = lane & 15;
    int khi = lane >> 4;                     /* 0: K={0,1}, 1: K={2,3} */
    int rbase = khi << 3;                    /* lanes 16-31 hold rows +8 */

    const float* arow = A + (size_t)(m0 + l15) * D_MODEL;

    v8f acc[4];
#pragma unroll
    for (int c = 0; c < 4; ++c) acc[c] = (v8f){};

#pragma unroll 2
    for (int kb = 0; kb < D_MODEL; kb += 4) {
        v2f a = *(const v2f*)(arow + kb + 2 * khi);
        const float* wp = W + (size_t)(kb + 2 * khi) * D_MODEL + n0 + l15;
#pragma unroll
        for (int c = 0; c < 4; ++c) {
            v2f b; b.x = wp[c * 16]; b.y = wp[D_MODEL + c * 16];
            acc[c] = __builtin_amdgcn_wmma_f32_16x16x4_f32(
                false, a, false, b, (short)0, acc[c], false, false);
        }
    }

    if (!split_heads) {
#pragma unroll
        for (int c = 0; c < 4; ++c) {
            float bv = bias[n0 + c * 16 + l15];
#pragma unroll
            for (int r = 0; r < 8; ++r) {
                int row = m0 + rbase + r;
                C[(size_t)row * D_MODEL + n0 + c * 16 + l15] = acc[c][r] + bv;
            }
        }
    } else {
        int h = n0 >> 6;                     /* 64-wide tile => one head */
#pragma unroll
        for (int c = 0; c < 4; ++c) {
            int d = c * 16 + l15;
            float bv = bias[n0 + d];
#pragma unroll
            for (int r = 0; r < 8; ++r) {
                int row = m0 + rbase + r;    /* in [0, B*S) */
                int bi = row / SEQ, s = row % SEQ;
                C[(((size_t)bi * NHEAD + h) * SEQ + s) * DK + d] = acc[c][r] + bv;
            }
        }
    }
}

// ---------------------------------------------------------------------------
// Attention core: one wave per (b, h, 16-row query tile).
// Q tile is held in 32 VGPRs for the whole key loop; stage 1 loads only K.
// Score tile bounces through LDS (C-layout -> A-layout transpose).
// ---------------------------------------------------------------------------
__global__ __launch_bounds__(256) void attn_kernel(
    const float* __restrict__ QP, const float* __restrict__ KP,
    const float* __restrict__ VP, const int* __restrict__ mask,
    float* __restrict__ X)
{
    __shared__ float smem[WPB][16 * SMSTRIDE];
    int wv   = (int)(threadIdx.x >> 5);
    int lane = (int)(threadIdx.x & 31);
    int wave = (int)blockIdx.x * WPB + wv;

    int qt = wave % (SEQ / 16);
    int bh = wave / (SEQ / 16);              /* [0, B*H) */
    int bi = bh / NHEAD, h = bh % NHEAD;
    int s0 = qt * 16;
    int l15 = lane & 15;
    int khi = lane >> 4;
    int rbase = khi << 3;

    const float* Q = QP + (size_t)bh * SEQ * DK;
    const float* K = KP + (size_t)bh * SEQ * DK;
    const float* V = VP + (size_t)bh * SEQ * DK;
    float* sm = smem[wv];

    /* ---- load Q tile once: per-lane footprint is 16 x float2 = 32 VGPRs */
    const float* qrow = Q + (size_t)(s0 + l15) * DK;
    v2f qv[16];
#pragma unroll
    for (int j = 0; j < 16; ++j)
        qv[j] = *(const v2f*)(qrow + 4 * j + 2 * khi);

    v8f acc[4];
#pragma unroll
    for (int c = 0; c < 4; ++c) acc[c] = (v8f){};

    for (int kb = 0; kb < SEQ; kb += 16) {
        /* ---- scores tile: 16x16, K-dim = 64 (16 WMMAs, K loads only) ---- */
        v8f sacc = {};
        const float* krow = K + (size_t)(kb + l15) * DK;
#pragma unroll
        for (int j = 0; j < 16; ++j) {
            v2f bb = *(const v2f*)(krow + 4 * j + 2 * khi);
            sacc = __builtin_amdgcn_wmma_f32_16x16x4_f32(
                false, qv[j], false, bb, (short)0, sacc, false, false);
        }

        /* ---- scale, mask, C-layout -> LDS (row-major, stride 18) ---- */
        int tg = kb + l15;
#pragma unroll
        for (int r = 0; r < 8; ++r) {
            int srow = rbase + r;                   /* 0..15 in tile */
            int sg = s0 + srow;
            float val = sacc[r] * 0.125f;           /* 1/sqrt(64) */
            if (mask[(size_t)sg * SEQ + tg] == 0) val = NEGV;
            sm[srow * SMSTRIDE + l15] = val;
        }
        asm volatile("s_wait_dscnt 0" ::: "memory");

        /* ---- out tile += scores(16x16) @ V_block(16x64), 16 WMMAs ---- */
#pragma unroll
        for (int j = 0; j < 4; ++j) {
            v2f a = *(const v2f*)(sm + l15 * SMSTRIDE + 4 * j + 2 * khi);
            const float* vrow = V + (size_t)(kb + 4 * j + 2 * khi) * DK + l15;
#pragma unroll
            for (int c = 0; c < 4; ++c) {
                v2f bb; bb.x = vrow[c * 16]; bb.y = vrow[DK + c * 16];
                acc[c] = __builtin_amdgcn_wmma_f32_16x16x4_f32(
                    false, a, false, bb, (short)0, acc[c], false, false);
            }
        }
    }

    /* ---- writeback merged-head layout [B, S, D_MODEL] ---- */
#pragma unroll
    for (int c = 0; c < 4; ++c) {
#pragma unroll
        for (int r = 0; r < 8; ++r) {
            int s = s0 + rbase + r;
            int d = c * 16 + l15;
            X[((size_t)bi * SEQ + s) * D_MODEL + h * DK + d] = acc[c][r];
        }
    }
}

// ---------------------------------------------------------------------------
extern "C" void kernel_launch(void* const* d_in, const int* in_sizes, int n_in,
                              void* d_out, int out_size, void* d_ws, size_t ws_size,
                              hipStream_t stream) {
    (void)in_sizes; (void)n_in; (void)out_size; (void)ws_size;
    const float* q    = (const float*)d_in[0];
    const float* k    = (const float*)d_in[1];
    const float* v    = (const float*)d_in[2];
    const int*   mask = (const int*)d_in[3];
    const float* w_q  = (const float*)d_in[4];
    const float* b_q  = (const float*)d_in[5];
    const float* w_k  = (const float*)d_in[6];
    const float* b_k  = (const float*)d_in[7];
    const float* w_v  = (const float*)d_in[8];
    const float* b_v  = (const float*)d_in[9];
    const float* w_o  = (const float*)d_in[10];
    const float* b_o  = (const float*)d_in[11];

    float* ws = (float*)d_ws;
    size_t proj = (size_t)M_ROWS * D_MODEL;
    float* qp = ws;
    float* kp = ws + proj;
    float* vp = ws + 2 * proj;
    float* xb = ws + 3 * proj;

    dim3 blk(256);
    int gemm_waves = (M_ROWS / 16) * (D_MODEL / 64);   /* 3072 */
    dim3 g_gemm(gemm_waves / WPB);
    int attn_waves = BATCH * NHEAD * (SEQ / 16);       /* 3072 */
    dim3 g_attn(attn_waves / WPB);

    gemm768_kernel<<<g_gemm, blk, 0, stream>>>(q, w_q, b_q, qp, 1);
    gemm768_kernel<<<g_gemm, blk, 0, stream>>>(k, w_k, b_k, kp, 1);
    gemm768_kernel<<<g_gemm, blk, 0, stream>>>(v, w_v, b_v, vp, 1);
    attn_kernel<<<g_attn, blk, 0, stream>>>(qp, kp, vp, mask, xb);
    gemm768_kernel<<<g_gemm, blk, 0, stream>>>(xb, w_o, b_o, (float*)d_out, 0);
}